// SUFGConv_12309376270507
// MI455X (gfx1250) — compile-verified
//
#include <hip/hip_runtime.h>
#include <cmath>

typedef float v2f __attribute__((ext_vector_type(2)));
typedef float v8f __attribute__((ext_vector_type(8)));

#define WMMA_F32(A, Bf, C) \
    __builtin_amdgcn_wmma_f32_16x16x4_f32(false, (A), false, (Bf), (short)0, (C), false, false)

// ---------------------------------------------------------------------------
// Sizes (compile-time constants from the reference)
// ---------------------------------------------------------------------------
#define BB       2
#define NN       2048
#define FF       32
#define DROWS    8192            // R*LEV*N
#define DCOLS    2049            // N+1
#define MSHR     6144            // rows of x_shrink (d_list rows 2048..8191)
#define KOUT     2049            // output rows per batch
#define KPAD     2064            // 129 tiles * 16
#define NSPLIT   4               // K-split for pass 3
#define KSEG     (MSHR / NSPLIT) // 1536

// Workspace layout (floats)
#define WS_XW_OFF    0                       // 2*2048*32   = 131072
#define WS_XSHR_OFF  (BB * NN * FF)          // 2*6144*32   = 393216
#define WS_PART_OFF  (WS_XSHR_OFF + BB * MSHR * FF)
// partial: BB * NSPLIT * KPAD * 32 = 528384 floats

// ---------------------------------------------------------------------------
// Pass 1: xw[b] = x[b] (2048x32) @ weight (32x32)   -- tiny, plain FMA
// ---------------------------------------------------------------------------
__global__ void sufg_xw_kernel(const float* __restrict__ x,
                               const float* __restrict__ w,
                               float* __restrict__ xw) {
    int idx = blockIdx.x * blockDim.x + threadIdx.x;   // [0, 2*2048*32)
    int o = idx & 31;
    int n = (idx >> 5) & (NN - 1);
    int b = idx >> 16;                                  // 2048*32 == 1<<16
    const float* xr = x + ((size_t)b * NN + n) * FF;
    float s = 0.f;
#pragma unroll
    for (int f = 0; f < FF; ++f) s = fmaf(xr[f], w[f * FF + o], s);
    xw[idx] = s;
}

// ---------------------------------------------------------------------------
// Pass 2: x_shrink[b] = shrink( filt * (d_list[b, 2048:, 1:] @ xw[b]) )
// One wave per 16-row tile, 16x32 output tile, K = 2048 via WMMA f32 16x16x4.
// A starts at an odd float offset (col 1) -> scalar b32 loads for A.
// ---------------------------------------------------------------------------
__global__ void __launch_bounds__(256)
sufg_pass2_kernel(const float* __restrict__ dl,
                  const float* __restrict__ xw,
                  const float* __restrict__ filt,
                  float* __restrict__ xshr,
                  float thr) {
    const int lane = threadIdx.x & 31;
    const int wave = threadIdx.x >> 5;
    const int tile = blockIdx.x * 8 + wave;       // 0..383
    const int b    = blockIdx.y;
    if (tile >= MSHR / 16) return;

    const int m0   = tile * 16;
    const int row  = lane & 15;
    const int hi   = lane >> 4;
    const int koff = hi << 1;                     // 0 or 2

    // A: y row = d_list[b, 2048+m0+row, 1+k+koff]
    const float* A  = dl + (size_t)b * DROWS * DCOLS
                         + (size_t)(2048 + m0 + row) * DCOLS + 1 + koff;
    // B: xw[b][(k+koff)..][col], col = lane&15 (tile0) / +16 (tile1)
    const float* Bp = xw + (size_t)b * NN * FF + (size_t)koff * FF + row;

    v8f acc0 = {0.f,0.f,0.f,0.f,0.f,0.f,0.f,0.f};
    v8f acc1 = {0.f,0.f,0.f,0.f,0.f,0.f,0.f,0.f};

    for (int kc = 0; kc < NN; kc += 64) {
        __builtin_prefetch(A + 256, 0, 0);        // ~1KB ahead along this row
#pragma unroll
        for (int k = 0; k < 64; k += 4) {
            v2f a;  a.x  = A[0];   a.y  = A[1];
            v2f b0; b0.x = Bp[0];  b0.y = Bp[32];
            v2f b1; b1.x = Bp[16]; b1.y = Bp[48];
            acc0 = WMMA_F32(a, b0, acc0);
            acc1 = WMMA_F32(a, b1, acc1);
            A += 4; Bp += 4 * FF;
        }
    }

    // Epilogue: filt scale, soft-threshold for high rows, store x_shrink
#pragma unroll
    for (int r = 0; r < 8; ++r) {
        const int m  = m0 + r + hi * 8;           // x_shrink row (0..6143)
        const float sc = filt[2048 + m];
        float v0 = acc0[r] * sc;
        float v1 = acc1[r] * sc;
        if (m >= 2048) {                          // "high" rows: soft threshold
            v0 = copysignf(fmaxf(fabsf(v0) - thr, 0.f), v0);
            v1 = copysignf(fmaxf(fabsf(v1) - thr, 0.f), v1);
        }
        float* orow = xshr + ((size_t)b * MSHR + m) * FF;
        orow[row]      = v0;
        orow[16 + row] = v1;
    }
}

// ---------------------------------------------------------------------------
// Pass 3: partial[b,s] = y2_seg @ x_shrink_seg, y2 = flat view of
// d_list[b, 2048:, :] as (2049 x 6144) row-major. K split 4-way (determinism
// via workspace partials, no float atomics). A here is 8B-aligned -> float2.
// ---------------------------------------------------------------------------
__global__ void __launch_bounds__(256)
sufg_pass3_kernel(const float* __restrict__ dl,
                  const float* __restrict__ xshr,
                  float* __restrict__ partial) {
    const int lane = threadIdx.x & 31;
    const int wave = threadIdx.x >> 5;
    const int unit = blockIdx.x * 8 + wave;       // 0..515
    const int b    = blockIdx.y;
    if (unit >= (KPAD / 16) * NSPLIT) return;

    const int split  = unit & (NSPLIT - 1);
    const int tile   = unit >> 2;                 // 0..128
    const int kbegin = split * KSEG;

    const int row  = lane & 15;
    const int hi   = lane >> 4;
    const int koff = hi << 1;

    int krow = tile * 16 + row;                   // output row this lane loads
    if (krow > KOUT - 1) krow = KOUT - 1;         // clamp (rows 2049..2063 unused)

    const float* A  = dl + (size_t)b * DROWS * DCOLS + (size_t)2048 * DCOLS
                         + (size_t)krow * MSHR + kbegin + koff;
    const float* Bp = xshr + (size_t)b * MSHR * FF + (size_t)(kbegin + koff) * FF + row;

    v8f acc0 = {0.f,0.f,0.f,0.f,0.f,0.f,0.f,0.f};
    v8f acc1 = {0.f,0.f,0.f,0.f,0.f,0.f,0.f,0.f};

    for (int kc = 0; kc < KSEG; kc += 64) {
        __builtin_prefetch(A + 256, 0, 0);
#pragma unroll
        for (int k = 0; k < 64; k += 4) {
            v2f a  = *(const v2f*)A;              // 8B-aligned here
            v2f b0; b0.x = Bp[0];  b0.y = Bp[32];
            v2f b1; b1.x = Bp[16]; b1.y = Bp[48];
            acc0 = WMMA_F32(a, b0, acc0);
            acc1 = WMMA_F32(a, b1, acc1);
            A += 4; Bp += 4 * FF;
        }
    }

#pragma unroll
    for (int r = 0; r < 8; ++r) {
        const int kk = tile * 16 + r + hi * 8;    // < KPAD always
        float* prow = partial + (((size_t)(b * NSPLIT + split)) * KPAD + kk) * FF;
        prow[row]      = acc0[r];
        prow[16 + row] = acc1[r];
    }
}

// ---------------------------------------------------------------------------
// Pass 4: out[b,k,o] = bias[o] + sum_s partial[b,s,k,o]   (k < 2049)
// ---------------------------------------------------------------------------
__global__ void sufg_reduce_kernel(const float* __restrict__ partial,
                                   const float* __restrict__ bias,
                                   float* __restrict__ out) {
    int idx = blockIdx.x * blockDim.x + threadIdx.x;
    if (idx >= BB * KOUT * FF) return;
    int o = idx & 31;
    int k = (idx >> 5) % KOUT;
    int b = idx / (KOUT * FF);
    float s = bias[o];
#pragma unroll
    for (int sp = 0; sp < NSPLIT; ++sp)
        s += partial[(((size_t)(b * NSPLIT + sp)) * KPAD + k) * FF + o];
    out[idx] = s;
}

// ---------------------------------------------------------------------------
extern "C" void kernel_launch(void* const* d_in, const int* in_sizes, int n_in,
                              void* d_out, int out_size, void* d_ws, size_t ws_size,
                              hipStream_t stream) {
    (void)in_sizes; (void)n_in; (void)out_size; (void)ws_size;
    const float* x    = (const float*)d_in[0];   // (2, 2048, 32)
    const float* dl   = (const float*)d_in[1];   // (2, 8192, 2049)
    const float* w    = (const float*)d_in[2];   // (32, 32)
    const float* filt = (const float*)d_in[3];   // (8192, 1)
    const float* bias = (const float*)d_in[4];   // (32,)
    float* out = (float*)d_out;                  // (2, 2049, 32)

    float* ws      = (float*)d_ws;
    float* xw      = ws + WS_XW_OFF;             // 2*2048*32
    float* xshr    = ws + WS_XSHR_OFF;           // 2*6144*32
    float* partial = ws + WS_PART_OFF;           // 2*4*2064*32

    const float thr = sqrtf(2.0f * logf((float)NN)) / sqrtf((float)NN); // SIGMA=1

    sufg_xw_kernel<<<(BB * NN * FF) / 256, 256, 0, stream>>>(x, w, xw);
    sufg_pass2_kernel<<<dim3((MSHR / 16) / 8, BB), 256, 0, stream>>>(dl, xw, filt, xshr, thr);
    sufg_pass3_kernel<<<dim3(((KPAD / 16) * NSPLIT + 7) / 8, BB), 256, 0, stream>>>(dl, xshr, partial);
    sufg_reduce_kernel<<<(BB * KOUT * FF + 255) / 256, 256, 0, stream>>>(partial, bias, out);
}